// MessagePassingLayer_18614388260936
// MI455X (gfx1250) — compile-verified
//
#include <hip/hip_runtime.h>
#include <cstddef>

#define NN 50000
#define DD 128
#define EE 800000

typedef float v2f __attribute__((ext_vector_type(2)));
typedef float v8f __attribute__((ext_vector_type(8)));

// ---------------------------------------------------------------------------
// utility
// ---------------------------------------------------------------------------
__global__ void fill_f32(float* __restrict__ p, float v, int n) {
  int i = blockIdx.x * blockDim.x + threadIdx.x;
  if (i < n) p[i] = v;
}

// ---------------------------------------------------------------------------
// GEMM: out = A(50000x128) @ W(128x128) + bias, fp32 WMMA 16x16x4.
// One block = 16 rows; 8 waves cover 8 column tiles of 16. A tile in LDS
// (row stride 132 floats -> conflict-free column reads).
// ---------------------------------------------------------------------------
__global__ __launch_bounds__(256) void gemm_bias_wmma(
    const float* __restrict__ A, const float* __restrict__ W,
    const float* __restrict__ bias, float* __restrict__ out)
{
  __shared__ float As[16][132];
  const int tid = threadIdx.x;
  const int tm  = blockIdx.x << 4;

  for (int idx = tid; idx < 16 * 128; idx += 256)
    As[idx >> 7][idx & 127] = A[(size_t)(tm + (idx >> 7)) * DD + (idx & 127)];
  __syncthreads();

  const int lane  = tid & 31;
  const int wave  = tid >> 5;
  const int tn    = wave << 4;            // column tile base (0..112)
  const int m     = lane & 15;            // A-frag row / C col index
  const int nIdx  = tn + m;               // output column
  const int khalf = (lane >> 4) << 1;     // K sub-offset per half-wave

  v8f c;
  const float bv = bias[nIdx];
  #pragma unroll
  for (int r = 0; r < 8; ++r) c[r] = bv;

  #pragma unroll
  for (int k = 0; k < DD; k += 4) {
    v2f a, b;
    a.x = As[m][k + khalf];
    a.y = As[m][k + khalf + 1];
    b.x = W[(size_t)(k + khalf)     * DD + nIdx];
    b.y = W[(size_t)(k + khalf + 1) * DD + nIdx];
    c = __builtin_amdgcn_wmma_f32_16x16x4_f32(false, a, false, b,
                                              (short)0, c, false, false);
  }

  const int mhi = (lane >> 4) << 3;       // M offset 0 or 8
  #pragma unroll
  for (int r = 0; r < 8; ++r)
    out[(size_t)(tm + mhi + r) * DD + nIdx] = c[r];
}

// ---------------------------------------------------------------------------
// per-edge-thread kernels
// ---------------------------------------------------------------------------
__global__ void deg_kernel(const int* __restrict__ row,
                           const float* __restrict__ emask,
                           float* __restrict__ deg) {
  int e = blockIdx.x * blockDim.x + threadIdx.x;
  if (e >= EE) return;
  float em = emask[e];
  if (em != 0.f) atomicAdd(&deg[row[e]], em);
}

__global__ void gcn_norm_kernel(const int* __restrict__ row,
                                const int* __restrict__ col,
                                const float* __restrict__ emask,
                                const float* __restrict__ deg,
                                float* __restrict__ normc) {
  int e = blockIdx.x * blockDim.x + threadIdx.x;
  if (e >= EE) return;
  float em = emask[e];
  float dr = deg[row[e]], dc = deg[col[e]];
  float nr = (dr > 0.f) ? rsqrtf(dr) : 0.f;
  float nc = (dc > 0.f) ? rsqrtf(dc) : 0.f;
  normc[e] = nr * nc * em;
}

__global__ void calew1_kernel(const float* __restrict__ weights,
                              const int* __restrict__ row,
                              const int* __restrict__ col,
                              const float* __restrict__ emask,
                              const float* __restrict__ deg,
                              float* __restrict__ w_send,
                              float* __restrict__ aggr) {
  int e = blockIdx.x * blockDim.x + threadIdx.x;
  if (e >= EE) return;
  int r = row[e];
  float d  = deg[r];
  float ws = ((d > 0.f) ? (weights[r] / d) : 0.f) * emask[e];
  w_send[e] = ws;
  if (ws != 0.f) atomicAdd(&aggr[col[e]], ws);
}

__global__ void calew_fin_kernel(float* __restrict__ aggr,
                                 float* __restrict__ weights) {
  int n = blockIdx.x * blockDim.x + threadIdx.x;
  if (n >= NN) return;
  float a = aggr[n] + 1e-12f;
  aggr[n] = a;
  weights[n] = a;
}

__global__ void calew2_kernel(const float* __restrict__ w_send,
                              const int* __restrict__ col,
                              const float* __restrict__ aggr,
                              float* __restrict__ ec) {
  int e = blockIdx.x * blockDim.x + threadIdx.x;
  if (e >= EE) return;
  ec[e] = w_send[e] / aggr[col[e]];
}

__global__ void filter_adj_kernel(const float* __restrict__ em_in,
                                  const int* __restrict__ row,
                                  const int* __restrict__ col,
                                  const float* __restrict__ kept,
                                  float* __restrict__ em_out) {
  int e = blockIdx.x * blockDim.x + threadIdx.x;
  if (e >= EE) return;
  em_out[e] = em_in[e] * kept[row[e]] * kept[col[e]];
}

// ---------------------------------------------------------------------------
// wave-per-edge scatter: out[dst[e], :] += coef[e] * h[src[e], :]
// ---------------------------------------------------------------------------
__global__ __launch_bounds__(256) void edge_scatter(
    const float* __restrict__ h, const int* __restrict__ src,
    const int* __restrict__ dst, const float* __restrict__ coef,
    float* __restrict__ out)
{
  int e    = (blockIdx.x * blockDim.x + threadIdx.x) >> 5;
  int lane = threadIdx.x & 31;
  if (e >= EE) return;
  float cf = coef[e];
  if (cf == 0.f) return;                       // wave-uniform
  float4 v = ((const float4*)(h + (size_t)src[e] * DD))[lane];
  float* o = out + (size_t)dst[e] * DD + (lane << 2);
  atomicAdd(o + 0, cf * v.x);
  atomicAdd(o + 1, cf * v.y);
  atomicAdd(o + 2, cf * v.z);
  atomicAdd(o + 3, cf * v.w);
}

// ---------------------------------------------------------------------------
// wave-per-node: score[n] = dot(h[n], p) / ||p||
// ---------------------------------------------------------------------------
__global__ __launch_bounds__(256) void score_kernel(
    const float* __restrict__ h, const float* __restrict__ p,
    float* __restrict__ score)
{
  int n    = (blockIdx.x * blockDim.x + threadIdx.x) >> 5;
  int lane = threadIdx.x & 31;
  if (n >= NN) return;
  float4 hv = ((const float4*)(h + (size_t)n * DD))[lane];
  float4 pv = ((const float4*)p)[lane];
  float s  = hv.x * pv.x + hv.y * pv.y + hv.z * pv.z + hv.w * pv.w;
  float pp = pv.x * pv.x + pv.y * pv.y + pv.z * pv.z + pv.w * pv.w;
  #pragma unroll
  for (int off = 16; off > 0; off >>= 1) {
    s  += __shfl_xor(s,  off, 32);
    pp += __shfl_xor(pp, off, 32);
  }
  if (lane == 0) score[n] = s * rsqrtf(pp);
}

// ---------------------------------------------------------------------------
// order-preserving key for float compare (handles -inf masking)
// ---------------------------------------------------------------------------
__device__ __forceinline__ unsigned float_key(float f) {
  unsigned b = __float_as_uint(f);
  return (b & 0x80000000u) ? ~b : (b | 0x80000000u);
}

// single-block 32-iteration radix bisection: largest t with |{key>=t}| >= k
__global__ __launch_bounds__(1024) void topk_threshold(
    const float* __restrict__ score, const float* __restrict__ mask,
    int k, unsigned* __restrict__ thr)
{
  __shared__ unsigned s_lo, s_hi;
  __shared__ int red[32];
  const int tid = threadIdx.x;
  if (tid == 0) { s_lo = 0u; s_hi = 0xFFFFFFFFu; }
  for (int it = 0; it < 32; ++it) {
    __syncthreads();
    unsigned lo = s_lo, hi = s_hi;
    unsigned d = hi - lo;
    unsigned mid = lo + (d >> 1) + (d & 1u);   // in (lo, hi], overflow-safe
    int cnt = 0;
    for (int n = tid; n < NN; n += 1024) {
      unsigned key = (mask[n] > 0.f) ? float_key(score[n]) : 0u;
      cnt += (key >= mid) ? 1 : 0;
    }
    #pragma unroll
    for (int off = 16; off > 0; off >>= 1) cnt += __shfl_xor(cnt, off, 32);
    if ((tid & 31) == 0) red[tid >> 5] = cnt;
    __syncthreads();
    if (tid == 0) {
      int tot = 0;
      for (int wv = 0; wv < 32; ++wv) tot += red[wv];
      if (tot >= k) s_lo = mid; else s_hi = mid - 1u;
    }
  }
  __syncthreads();
  if (tid == 0) *thr = s_lo;
}

// wave-per-node: kept mask, x = h * tanh(score) * kept, update node state
__global__ __launch_bounds__(256) void apply_topk(
    const float* __restrict__ h, const float* __restrict__ score,
    const unsigned* __restrict__ thr, float* __restrict__ x,
    float* __restrict__ node_mask, float* __restrict__ weights)
{
  int n    = (blockIdx.x * blockDim.x + threadIdx.x) >> 5;
  int lane = threadIdx.x & 31;
  if (n >= NN) return;
  float sc = score[n];
  unsigned key = (node_mask[n] > 0.f) ? float_key(sc) : 0u;
  float kept = (key >= *thr) ? 1.f : 0.f;
  float scale = tanhf(sc) * kept;
  float4 v = ((const float4*)(h + (size_t)n * DD))[lane];
  float4 o; o.x = v.x * scale; o.y = v.y * scale; o.z = v.z * scale; o.w = v.w * scale;
  ((float4*)(x + (size_t)n * DD))[lane] = o;
  if (lane == 0) { node_mask[n] = kept; weights[n] *= kept; }
}

__global__ void add_kernel(const float* __restrict__ a,
                           const float* __restrict__ b,
                           float* __restrict__ o, int n) {
  int i = blockIdx.x * blockDim.x + threadIdx.x;
  if (i < n) o[i] = a[i] + b[i];
}

// ---------------------------------------------------------------------------
// orchestration
// ---------------------------------------------------------------------------
extern "C" void kernel_launch(void* const* d_in, const int* in_sizes, int n_in,
                              void* d_out, int out_size, void* d_ws, size_t ws_size,
                              hipStream_t stream)
{
  (void)in_sizes; (void)n_in; (void)out_size; (void)ws_size;
  const float* x_in = (const float*)d_in[0];
  const int*   ei   = (const int*)  d_in[1];
  const float* Wd   = (const float*)d_in[2];
  const float* bd   = (const float*)d_in[3];
  const float* Wu   = (const float*)d_in[4];
  const float* bu   = (const float*)d_in[5];
  const float* Wb   = (const float*)d_in[6];
  const float* bb   = (const float*)d_in[7];
  const float* pv   = (const float*)d_in[8];
  const int* row = ei;
  const int* col = ei + EE;

  // ---- workspace carving (~150 MB; everything fits global L2) ----
  char* wsb = (char*)d_ws;
  size_t ofs = 0;
  auto carve = [&](size_t nf) -> float* {
    float* p = (float*)(wsb + ofs);
    ofs += ((nf * sizeof(float) + 255) & ~(size_t)255);
    return p;
  };
  const size_t ND = (size_t)NN * DD;
  float* T1    = carve(ND);   // gemm output scratch
  float* T2    = carve(ND);   // conv stream buffer
  float* XB    = carve(ND);   // current x
  float* D0    = carve(ND);   // down_outs[0]
  float* D1    = carve(ND);   // down_outs[1]
  float* EH0   = carve(EE);   // emask hist level0 (all ones)
  float* EH1   = carve(EE);   // emask hist level1
  float* EM2   = carve(EE);   // emask for bottom
  float* CT0   = carve(EE);   // cts[0]
  float* CT1   = carve(EE);   // cts[1]
  float* WSND  = carve(EE);   // w_send scratch
  float* NORMC = carve(EE);   // per-edge gcn norm
  float* DEG0  = carve(NN);
  float* DEG1  = carve(NN);
  float* DEG2  = carve(NN);
  float* AGGR  = carve(NN);
  float* WTS   = carve(NN);
  float* NMASK = carve(NN);
  float* SCORE = carve(NN);
  unsigned* THRESH = (unsigned*)carve(64);

  const dim3 B256(256), B1024(1024);
  const dim3 gEdgeT((EE + 255) / 256);
  const dim3 gNodeT((NN + 255) / 256);
  const dim3 gEdgeW((EE * 32) / 256);        // wave per edge
  const dim3 gNodeW((NN * 32 + 255) / 256);  // wave per node
  const dim3 gGemm(NN / 16);                 // 3125 row tiles (exact)
  const dim3 gND(((int)ND + 255) / 256);

  auto fill = [&](float* p, float v, int n) {
    fill_f32<<<dim3((n + 255) / 256), B256, 0, stream>>>(p, v, n);
  };
  // GCN conv: gemm(in)->T1, zero(out), scatter norm*T1[row] -> out[col].
  // (out == in is safe: gemm fully consumes `in` first.)
  auto gcn = [&](const float* in, const float* Wp, const float* bp, float* out) {
    gemm_bias_wmma<<<gGemm, B256, 0, stream>>>(in, Wp, bp, T1);
    fill(out, 0.f, (int)ND);
    edge_scatter<<<gEdgeW, B256, 0, stream>>>(T1, row, col, NORMC, out);
  };
  const size_t DSQ = (size_t)DD * DD;

  // ---- init graph state ----
  fill(EH0, 1.f, EE);
  fill(WTS, 1.f, NN);
  fill(NMASK, 1.f, NN);

  const int kvals[2] = {25000, 12500};   // ceil(0.5*50000), ceil(0.5*25000)
  float* EMH[2]  = {EH0, EH1};
  float* EMNXT[2]= {EH1, EM2};
  float* DEGL[2] = {DEG0, DEG1};
  float* CTL[2]  = {CT0, CT1};
  float* DNS[2]  = {D0, D1};

  // ======== down path ========
  for (int i = 0; i < 2; ++i) {
    float* em  = EMH[i];
    float* dg  = DEGL[i];
    fill(dg, 0.f, NN);
    deg_kernel<<<gEdgeT, B256, 0, stream>>>(row, em, dg);
    gcn_norm_kernel<<<gEdgeT, B256, 0, stream>>>(row, col, em, dg, NORMC);

    const float* in0 = (i == 0) ? x_in : XB;
    gcn(in0, Wd + (size_t)(i * 2 + 0) * DSQ, bd + (size_t)(i * 2 + 0) * DD, T2);
    gcn(T2,  Wd + (size_t)(i * 2 + 1) * DSQ, bd + (size_t)(i * 2 + 1) * DD, DNS[i]);

    // cal_ew (reuses dg = seg(emask,row))
    fill(AGGR, 0.f, NN);
    calew1_kernel<<<gEdgeT, B256, 0, stream>>>(WTS, row, col, em, dg, WSND, AGGR);
    calew_fin_kernel<<<gNodeT, B256, 0, stream>>>(AGGR, WTS);
    calew2_kernel<<<gEdgeT, B256, 0, stream>>>(WSND, col, AGGR, CTL[i]);

    // WeightedEdgeConv aggregating=True: h = seg(ew * h[row], col)
    fill(T2, 0.f, (int)ND);
    edge_scatter<<<gEdgeW, B256, 0, stream>>>(DNS[i], row, col, CTL[i], T2);

    // TopK pooling
    score_kernel<<<gNodeW, B256, 0, stream>>>(T2, pv + (size_t)i * DD, SCORE);
    topk_threshold<<<dim3(1), B1024, 0, stream>>>(SCORE, NMASK, kvals[i], THRESH);
    apply_topk<<<gNodeW, B256, 0, stream>>>(T2, SCORE, THRESH, XB, NMASK, WTS);
    filter_adj_kernel<<<gEdgeT, B256, 0, stream>>>(em, row, col, NMASK, EMNXT[i]);
  }

  // ======== bottom ========
  fill(DEG2, 0.f, NN);
  deg_kernel<<<gEdgeT, B256, 0, stream>>>(row, EM2, DEG2);
  gcn_norm_kernel<<<gEdgeT, B256, 0, stream>>>(row, col, EM2, DEG2, NORMC);
  gcn(XB, Wb, bb, T2);
  gcn(T2, Wb + DSQ, bb + DD, XB);

  // ======== up path ========
  for (int i = 0; i < 2; ++i) {
    int up = 1 - i;
    // WeightedEdgeConv aggregating=False: h = seg(ct * x[col], row)
    fill(T2, 0.f, (int)ND);
    edge_scatter<<<gEdgeW, B256, 0, stream>>>(XB, col, row, CTL[up], T2);

    gcn_norm_kernel<<<gEdgeT, B256, 0, stream>>>(row, col, EMH[up], DEGL[up], NORMC);
    gcn(T2, Wu + (size_t)(i * 2 + 0) * DSQ, bu + (size_t)(i * 2 + 0) * DD, T2);
    gcn(T2, Wu + (size_t)(i * 2 + 1) * DSQ, bu + (size_t)(i * 2 + 1) * DD, T2);

    float* dst = (i == 1) ? (float*)d_out : XB;
    add_kernel<<<gND, B256, 0, stream>>>(T2, DNS[up], dst, (int)ND);
  }
}